// DeepGraphInfomaxLoss_317827579956
// MI455X (gfx1250) — compile-verified
//
#include <hip/hip_runtime.h>
#include <hip/hip_bf16.h>

#define N_NODES 100000
#define N_EDGES 1600000
#define DIM     64
#define EPS_DGI 1e-15f

typedef __attribute__((ext_vector_type(2))) float v2f;
typedef __attribute__((ext_vector_type(8))) float v8f;

// ---------------------------------------------------------------------------
// Degree: deg[c] += 1 per edge, then dis = rsqrt(deg + 1)
// ---------------------------------------------------------------------------
__global__ __launch_bounds__(256) void dgi_deg_count(const int* __restrict__ col,
                                                     float* __restrict__ deg, int e) {
    int t = blockIdx.x * blockDim.x + threadIdx.x;
    if (t < e) atomicAdd(&deg[col[t]], 1.0f);
}

__global__ __launch_bounds__(256) void dgi_deg_finish(float* __restrict__ deg, int n) {
    int t = blockIdx.x * blockDim.x + threadIdx.x;
    if (t < n) deg[t] = rsqrtf(deg[t] + 1.0f);
}

// ---------------------------------------------------------------------------
// GEMM  H = X[rows] @ W   (N x 64) @ (64 x 64), f32, via v_wmma_f32_16x16x4_f32
// One wave -> one 16-row tile, all 4 column tiles.
// W is staged into LDS pre-swizzled via async global->LDS element copies:
//   element (k,n) -> LDS float index (k>>1)*128 + n*2 + (k&1)
// so a B fragment {W[kk][n], W[kk+1][n]} (kk even) is one aligned b64 LDS load.
// A frag (16x4, 32-bit): lane L, vgpr j -> A[L%16][2*(L/16)+j]  (float2 load)
// B frag (4x16, 32-bit): lane L, vgpr j -> B[2*(L/16)+j][L%16]
// C/D (16x16, 32-bit):   lane L, vgpr j -> D[j + 8*(L/16)][L%16]
// ---------------------------------------------------------------------------
__global__ __launch_bounds__(256)
void dgi_gemm64(const float* __restrict__ X, const float* __restrict__ W,
                float* __restrict__ H, const int* __restrict__ perm,
                int n, int usePerm) {
    __shared__ float Wsz[DIM * DIM];

    // async-stage W into LDS with the swizzle applied (16 elements per thread)
    {
        unsigned ldsBase = (unsigned)(uintptr_t)&Wsz[0];   // low 32 bits == LDS byte offset
        #pragma unroll
        for (int it = 0; it < 16; ++it) {
            int idx = threadIdx.x + it * 256;              // 0..4095
            int k = idx >> 6;
            int nn = idx & 63;
            unsigned dstOff = ldsBase + (((k >> 1) * 128 + nn * 2 + (k & 1)) << 2);
            const float* src = W + idx;
            asm volatile("global_load_async_to_lds_b32 %0, %1, off"
                         :: "v"(dstOff), "v"(src) : "memory");
        }
        asm volatile("s_wait_asynccnt 0x0" ::: "memory");
    }
    __syncthreads();

    const int wave = threadIdx.x >> 5;
    const int lane = threadIdx.x & 31;
    const int tile = blockIdx.x * 8 + wave;
    const int rowBase = tile * 16;
    if (rowBase >= n) return;                  // wave-uniform; N % 16 == 0

    const int lm = lane & 15;                  // L % 16
    const int lh = lane >> 4;                  // L / 16
    const int r = rowBase + lm;
    const int srcRow = usePerm ? perm[r] : r;
    const float* __restrict__ xrow = X + (size_t)srcRow * DIM;

    v8f acc0 = {}, acc1 = {}, acc2 = {}, acc3 = {};

    #pragma unroll
    for (int k0 = 0; k0 < DIM; k0 += 4) {
        v2f a = *(const v2f*)(xrow + k0 + 2 * lh);
        const int kk = (k0 >> 1) + lh;                    // (k0 + 2*lh) / 2
        v2f b0 = *(const v2f*)&Wsz[kk * 128 + ( 0 + lm) * 2];
        v2f b1 = *(const v2f*)&Wsz[kk * 128 + (16 + lm) * 2];
        v2f b2 = *(const v2f*)&Wsz[kk * 128 + (32 + lm) * 2];
        v2f b3 = *(const v2f*)&Wsz[kk * 128 + (48 + lm) * 2];
        acc0 = __builtin_amdgcn_wmma_f32_16x16x4_f32(false, a, false, b0, (short)0, acc0, false, false);
        acc1 = __builtin_amdgcn_wmma_f32_16x16x4_f32(false, a, false, b1, (short)0, acc1, false, false);
        acc2 = __builtin_amdgcn_wmma_f32_16x16x4_f32(false, a, false, b2, (short)0, acc2, false, false);
        acc3 = __builtin_amdgcn_wmma_f32_16x16x4_f32(false, a, false, b3, (short)0, acc3, false, false);
    }

    #pragma unroll
    for (int j = 0; j < 8; ++j) {
        const size_t rowOff = (size_t)(rowBase + j + 8 * lh) * DIM;
        H[rowOff +  0 + lm] = acc0[j];
        H[rowOff + 16 + lm] = acc1[j];
        H[rowOff + 32 + lm] = acc2[j];
        H[rowOff + 48 + lm] = acc3[j];
    }
}

// ---------------------------------------------------------------------------
// Edge scatter: agg[col] += h[row] * dis[row]*dis[col].  One wave per edge;
// edge id forced into an SGPR so row/col/dis loads become scalar loads.
// Each lane owns a float2 slice of the 64-wide feature vector.
// ---------------------------------------------------------------------------
__global__ __launch_bounds__(256)
void dgi_edge_scatter(const float* __restrict__ H, const float* __restrict__ dis,
                      const int* __restrict__ row, const int* __restrict__ col,
                      float* __restrict__ agg, int e) {
    const int gid = blockIdx.x * blockDim.x + threadIdx.x;
    const int lane = threadIdx.x & 31;
    const int edge = __builtin_amdgcn_readfirstlane(gid >> 5);   // wave-uniform
    if (edge >= e) return;
    const int r = row[edge];
    const int c = col[edge];
    const float nrm = dis[r] * dis[c];
    v2f v = *(const v2f*)(H + (size_t)r * DIM + lane * 2);
    float* dst = agg + (size_t)c * DIM + lane * 2;
    atomicAdd(dst + 0, v.x * nrm);
    atomicAdd(dst + 1, v.y * nrm);
}

// ---------------------------------------------------------------------------
// Combine: Z = agg + H * dis^2 + bias  (optional ReLU).  Z may alias agg.
// float2 granularity: one thread -> 2 features.
// ---------------------------------------------------------------------------
__global__ __launch_bounds__(256)
void dgi_combine(const float* __restrict__ H, const float* __restrict__ agg,
                 const float* __restrict__ dis, const float* __restrict__ bias,
                 float* __restrict__ Z, int n, int doRelu) {
    const int i2 = blockIdx.x * blockDim.x + threadIdx.x;       // float2 index
    if (i2 >= n * (DIM / 2)) return;
    const int node = i2 >> 5;
    const int d2 = (i2 & 31) * 2;
    const float di = dis[node];
    const float s = di * di;
    v2f h = *(const v2f*)(H + (size_t)i2 * 2);
    v2f a = *(const v2f*)(agg + (size_t)i2 * 2);
    v2f b = *(const v2f*)(bias + d2);
    float z0 = a.x + h.x * s + b.x;
    float z1 = a.y + h.y * s + b.y;
    if (doRelu) { z0 = fmaxf(z0, 0.0f); z1 = fmaxf(z1, 0.0f); }
    v2f z; z.x = z0; z.y = z1;
    *(v2f*)(Z + (size_t)i2 * 2) = z;
}

// ---------------------------------------------------------------------------
// Column sums of pos_z (for the summary vector)
// ---------------------------------------------------------------------------
__global__ __launch_bounds__(256)
void dgi_colsum(const float* __restrict__ Z, float* __restrict__ colsum, int n) {
    const int d = threadIdx.x & 63;
    const int sub = threadIdx.x >> 6;           // 0..3
    const int nchunks = gridDim.x * 4;
    const int chunk = blockIdx.x * 4 + sub;
    float s = 0.0f;
    for (int r = chunk; r < n; r += nchunks) s += Z[(size_t)r * DIM + d];
    atomicAdd(&colsum[d], s);
}

// ---------------------------------------------------------------------------
// summary = sigmoid(colsum / N);  wsv = W_dgi @ summary    (1 block, 64 thr)
// ---------------------------------------------------------------------------
__global__ __launch_bounds__(64)
void dgi_make_ws(const float* __restrict__ colsum, const float* __restrict__ Wdgi,
                 float* __restrict__ wsv, int n) {
    __shared__ float s[DIM];
    const int d = threadIdx.x;
    s[d] = 1.0f / (1.0f + expf(-colsum[d] / (float)n));
    __syncthreads();
    float acc = 0.0f;
    #pragma unroll 8
    for (int k = 0; k < DIM; ++k) acc += Wdgi[d * DIM + k] * s[k];
    wsv[d] = acc;
}

// ---------------------------------------------------------------------------
// Loss: wave per node; lanes reduce the 64-dot via shfl; lane0 accumulates.
// accum[0] = sum -log(sigmoid(pos.ws)+eps); accum[1] = sum -log(1-sigmoid(neg.ws)+eps)
// ---------------------------------------------------------------------------
__global__ __launch_bounds__(256)
void dgi_loss(const float* __restrict__ posZ, const float* __restrict__ negZ,
              const float* __restrict__ wsv, float* __restrict__ accum, int n) {
    const int gid = blockIdx.x * blockDim.x + threadIdx.x;
    const int node = gid >> 5;
    const int lane = threadIdx.x & 31;
    if (node >= n) return;
    v2f w = *(const v2f*)(wsv + lane * 2);
    v2f p = *(const v2f*)(posZ + (size_t)node * DIM + lane * 2);
    v2f q = *(const v2f*)(negZ + (size_t)node * DIM + lane * 2);
    float dp = p.x * w.x + p.y * w.y;
    float dn = q.x * w.x + q.y * w.y;
    #pragma unroll
    for (int off = 16; off > 0; off >>= 1) {
        dp += __shfl_down(dp, off, 32);
        dn += __shfl_down(dn, off, 32);
    }
    if (lane == 0) {
        const float pp = 1.0f / (1.0f + expf(-dp));
        const float pn = 1.0f / (1.0f + expf(-dn));
        atomicAdd(&accum[0], -logf(pp + EPS_DGI));
        atomicAdd(&accum[1], -logf(1.0f - pn + EPS_DGI));
    }
}

__global__ void dgi_finalize(const float* __restrict__ accum, float* __restrict__ out, int n) {
    if (threadIdx.x == 0 && blockIdx.x == 0)
        out[0] = (accum[0] + accum[1]) / (float)n;
}

// ---------------------------------------------------------------------------
extern "C" void kernel_launch(void* const* d_in, const int* in_sizes, int n_in,
                              void* d_out, int out_size, void* d_ws, size_t ws_size,
                              hipStream_t stream) {
    const float* x     = (const float*)d_in[0];
    const float* W1    = (const float*)d_in[1];
    const float* b1    = (const float*)d_in[2];
    const float* W2    = (const float*)d_in[3];
    const float* b2    = (const float*)d_in[4];
    const float* Wdgi  = (const float*)d_in[5];
    const int*   eidx  = (const int*)d_in[6];
    const int*   perm  = (const int*)d_in[7];
    const int*   row   = eidx;
    const int*   col   = eidx + N_EDGES;
    float* out = (float*)d_out;

    // workspace layout
    const size_t NPAD = 100096;           // N rounded up
    const size_t FEAT = (size_t)N_NODES * DIM;
    float* dis    = (float*)d_ws;
    float* bufA   = dis + NPAD;           // GEMM output h
    float* bufB   = bufA + FEAT;          // agg / z (in-place combine)
    float* pos_z  = bufB + FEAT;
    float* neg_z  = pos_z + FEAT;
    float* colsum = neg_z + FEAT;         // 64
    float* wsv    = colsum + DIM;         // 64
    float* accum  = wsv + DIM;            // 2

    const int B256 = 256;
    const int gDeg  = (N_EDGES + B256 - 1) / B256;
    const int gNode = (N_NODES + B256 - 1) / B256;
    const int gHalf = (int)((FEAT / 2 + B256 - 1) / B256);
    const int gGemm = (N_NODES / 16 + 7) / 8;                 // 8 waves/block
    const int gScat = (int)(((size_t)N_EDGES * 32 + B256 - 1) / B256);
    const int gLoss = (int)(((size_t)N_NODES * 32 + B256 - 1) / B256);

    // degree / normalization
    hipMemsetAsync(dis, 0, N_NODES * sizeof(float), stream);
    dgi_deg_count<<<gDeg, B256, 0, stream>>>(col, dis, N_EDGES);
    dgi_deg_finish<<<gNode, B256, 0, stream>>>(dis, N_NODES);

    // ---- positive encoding ----
    dgi_gemm64<<<gGemm, B256, 0, stream>>>(x, W1, bufA, perm, N_NODES, 0);
    hipMemsetAsync(bufB, 0, FEAT * sizeof(float), stream);
    dgi_edge_scatter<<<gScat, B256, 0, stream>>>(bufA, dis, row, col, bufB, N_EDGES);
    dgi_combine<<<gHalf, B256, 0, stream>>>(bufA, bufB, dis, b1, bufB, N_NODES, 1);
    dgi_gemm64<<<gGemm, B256, 0, stream>>>(bufB, W2, bufA, perm, N_NODES, 0);
    hipMemsetAsync(bufB, 0, FEAT * sizeof(float), stream);
    dgi_edge_scatter<<<gScat, B256, 0, stream>>>(bufA, dis, row, col, bufB, N_EDGES);
    dgi_combine<<<gHalf, B256, 0, stream>>>(bufA, bufB, dis, b2, pos_z, N_NODES, 0);

    // ---- negative encoding (x rows permuted) ----
    dgi_gemm64<<<gGemm, B256, 0, stream>>>(x, W1, bufA, perm, N_NODES, 1);
    hipMemsetAsync(bufB, 0, FEAT * sizeof(float), stream);
    dgi_edge_scatter<<<gScat, B256, 0, stream>>>(bufA, dis, row, col, bufB, N_EDGES);
    dgi_combine<<<gHalf, B256, 0, stream>>>(bufA, bufB, dis, b1, bufB, N_NODES, 1);
    dgi_gemm64<<<gGemm, B256, 0, stream>>>(bufB, W2, bufA, perm, N_NODES, 0);
    hipMemsetAsync(bufB, 0, FEAT * sizeof(float), stream);
    dgi_edge_scatter<<<gScat, B256, 0, stream>>>(bufA, dis, row, col, bufB, N_EDGES);
    dgi_combine<<<gHalf, B256, 0, stream>>>(bufA, bufB, dis, b2, neg_z, N_NODES, 0);

    // ---- summary / discriminator / loss ----
    hipMemsetAsync(colsum, 0, (DIM + DIM + 4) * sizeof(float), stream);
    dgi_colsum<<<256, B256, 0, stream>>>(pos_z, colsum, N_NODES);
    dgi_make_ws<<<1, 64, 0, stream>>>(colsum, Wdgi, wsv, N_NODES);
    dgi_loss<<<gLoss, B256, 0, stream>>>(pos_z, neg_z, wsv, accum, N_NODES);
    dgi_finalize<<<1, 64, 0, stream>>>(accum, out, N_NODES);
}